// LightPowerLIRALayer_40939628265957
// MI455X (gfx1250) — compile-verified
//
#include <hip/hip_runtime.h>
#include <stdint.h>

// Problem constants (fixed by the reference's setup_inputs()).
#define N_USERS   512
#define N_ITEMS   20000
#define NNZ_TOTAL 4000000
// Each CSR row padded to a multiple of 8 slots (pad entries: v=0, col=0).
#define ROW_PAD   8
#define PAD_MAX   (NNZ_TOTAL + ROW_PAD * N_ITEMS)   // 4,160,000 slots max

typedef __attribute__((ext_vector_type(2))) float v2f;
typedef __attribute__((ext_vector_type(8))) float v8f;

// ---------------------------------------------------------------------------
// Phase 0: zero the per-row counters and the padded CSR arrays.
// ---------------------------------------------------------------------------
__global__ void lira_zero_counts(unsigned* __restrict__ counts, int n) {
  int i = blockIdx.x * blockDim.x + threadIdx.x;
  if (i < n) counts[i] = 0u;
}

__global__ void lira_zero_csr(float* __restrict__ csr_val,
                              int* __restrict__ csr_col) {
  int p = blockIdx.x * blockDim.x + threadIdx.x;
  if (p < PAD_MAX) {
    csr_val[p] = 0.0f;
    csr_col[p] = 0;
  }
}

// ---------------------------------------------------------------------------
// Phase 1: per-row nonzero counts.
// ---------------------------------------------------------------------------
__global__ void lira_hist(const int* __restrict__ S_indices,
                          unsigned* __restrict__ counts) {
  int k = blockIdx.x * blockDim.x + threadIdx.x;
  if (k < NNZ_TOTAL) {
    int row = S_indices[2 * k];  // S_indices[k][0] = row i
    atomicAdd(&counts[row], 1u);
  }
}

// ---------------------------------------------------------------------------
// Phase 2: exclusive scan of PADDED row lengths -> row_start / cursor.
// Single 1024-thread block; each thread owns a 20-element chunk.
// row_start[i+1]-row_start[i] is the padded length (multiple of 8).
// ---------------------------------------------------------------------------
__global__ __launch_bounds__(1024) void lira_scan(
    const unsigned* __restrict__ counts,
    unsigned* __restrict__ row_start,
    unsigned* __restrict__ cursor) {
  __shared__ unsigned sums[1024];
  const int t = threadIdx.x;
  const int CHUNK = (N_ITEMS + 1023) / 1024;  // 20
  int lo = t * CHUNK;
  int hi = lo + CHUNK;
  if (hi > N_ITEMS) hi = N_ITEMS;
  if (lo > N_ITEMS) lo = N_ITEMS;

  unsigned s = 0;
  for (int i = lo; i < hi; ++i)
    s += (counts[i] + (ROW_PAD - 1)) & ~(unsigned)(ROW_PAD - 1);
  sums[t] = s;
  __syncthreads();

  // Hillis-Steele inclusive scan over the 1024 partials.
  for (int off = 1; off < 1024; off <<= 1) {
    unsigned v = (t >= off) ? sums[t - off] : 0u;
    __syncthreads();
    sums[t] += v;
    __syncthreads();
  }

  unsigned run = (t == 0) ? 0u : sums[t - 1];
  for (int i = lo; i < hi; ++i) {
    unsigned plen = (counts[i] + (ROW_PAD - 1)) & ~(unsigned)(ROW_PAD - 1);
    row_start[i] = run;
    cursor[i]    = run;
    run += plen;
  }
  if (t == 1023) row_start[N_ITEMS] = sums[1023];
}

// ---------------------------------------------------------------------------
// Phase 3: scatter COO -> padded CSR (val/col). Padding slots keep v=0.
// ---------------------------------------------------------------------------
__global__ void lira_scatter(const float* __restrict__ S_values,
                             const int* __restrict__ S_indices,
                             unsigned* __restrict__ cursor,
                             float* __restrict__ csr_val,
                             int* __restrict__ csr_col) {
  int k = blockIdx.x * blockDim.x + threadIdx.x;
  if (k < NNZ_TOTAL) {
    int row = S_indices[2 * k];
    int col = S_indices[2 * k + 1];
    unsigned pos = atomicAdd(&cursor[row], 1u);
    csr_val[pos] = S_values[k];
    csr_col[pos] = col;
  }
}

// ---------------------------------------------------------------------------
// Phase 4: X [512 x 20000] -> Xt [20000 x 512] so each gathered item column
// is one contiguous 2 KB row (Xt = 41 MB, resident in the 192 MB L2).
// ---------------------------------------------------------------------------
__global__ __launch_bounds__(256) void lira_transpose(
    const float* __restrict__ X, float* __restrict__ Xt) {
  __shared__ float tile[32][33];
  const int j0 = blockIdx.x * 32;  // item dim (20000 = 625*32)
  const int u0 = blockIdx.y * 32;  // user dim (512 = 16*32)
  const int tx = threadIdx.x, ty = threadIdx.y;  // 32 x 8
#pragma unroll
  for (int r = 0; r < 32; r += 8)
    tile[ty + r][tx] = X[(size_t)(u0 + ty + r) * N_ITEMS + (j0 + tx)];
  __syncthreads();
#pragma unroll
  for (int r = 0; r < 32; r += 8)
    Xt[(size_t)(j0 + ty + r) * N_USERS + (u0 + tx)] = tile[tx][ty + r];
}

// ---------------------------------------------------------------------------
// Phase 5: per-item-row gathered reduction via V_WMMA_F32_16X16X4_F32.
//
// One workgroup (256 thr = 8 wave32) per item row i. Wave w owns users
// [w*64, w*64+64) as 4 WMMA M-tiles of 16 users. Per K-chunk of 4 nonzeros:
//   A[m=user][k] = Xt[j_k][user]   (16x4 f32: VGPR0 holds K0|K2 across the
//                                   lane halves, VGPR1 holds K1|K3)
//   B[k][n]      = v_k  for all n  (4x16 f32, same K0/K2 lane-half split)
//   D[m][n]     += sum_k A[m][k]*v_k   -- identical in every column n.
// Rows are padded to multiples of 8 with (v=0, j=0), so the loop has NO
// tail guards and runs 2 K-chunks (8 WMMAs) per iteration; same-accumulator
// WMMAs are 4 instructions apart (covers the WMMA->WMMA hazard window).
// D layout: lane 0 holds users M=0..7 in acc VGPRs 0..7 (N=0), lane 16
// holds M=8..15, so two lanes per tile write the 16 results. No atomics.
// ---------------------------------------------------------------------------
__global__ __launch_bounds__(256) void lira_spmm_wmma(
    const float* __restrict__ Xt,
    const float* __restrict__ csr_val,
    const int* __restrict__ csr_col,
    const unsigned* __restrict__ row_start,
    float* __restrict__ out) {
  const int i    = blockIdx.x;          // item row
  const int lane = threadIdx.x & 31;    // wave32
  const int wave = threadIdx.x >> 5;    // 0..7
  const bool hiHalf = lane >= 16;
  const int  u0  = wave * 64 + (lane & 15);  // this lane's first user

  const unsigned start = row_start[i];
  const unsigned end   = row_start[i + 1];  // padded, multiple of 8 apart

  v8f c0 = {}; v8f c1 = {}; v8f c2 = {}; v8f c3 = {};

  for (unsigned k0 = start; k0 < end; k0 += 8) {
    // Wave-uniform CSR reads; all 8 slots are valid (padding is v=0,j=0).
    float v0 = csr_val[k0 + 0]; int j0 = csr_col[k0 + 0];
    float v1 = csr_val[k0 + 1]; int j1 = csr_col[k0 + 1];
    float v2 = csr_val[k0 + 2]; int j2 = csr_col[k0 + 2];
    float v3 = csr_val[k0 + 3]; int j3 = csr_col[k0 + 3];
    float v4 = csr_val[k0 + 4]; int j4 = csr_col[k0 + 4];
    float v5 = csr_val[k0 + 5]; int j5 = csr_col[k0 + 5];
    float v6 = csr_val[k0 + 6]; int j6 = csr_col[k0 + 6];
    float v7 = csr_val[k0 + 7]; int j7 = csr_col[k0 + 7];

    // Stream-prefetch the CSR arrays (global_prefetch_b8 on gfx1250).
    __builtin_prefetch(csr_val + k0 + 64, 0, 1);
    __builtin_prefetch(csr_col + k0 + 64, 0, 1);

    // ---- K-chunk A: nonzeros k0..k0+3 ----
    v2f b;
    b.x = hiHalf ? v2 : v0;   // K0 (lanes 0-15) / K2 (lanes 16-31)
    b.y = hiHalf ? v3 : v1;   // K1 / K3
    const float* ra = Xt + (size_t)(hiHalf ? j2 : j0) * N_USERS;
    const float* rb = Xt + (size_t)(hiHalf ? j3 : j1) * N_USERS;

    v2f a;
    a.x = ra[u0];      a.y = rb[u0];
    c0 = __builtin_amdgcn_wmma_f32_16x16x4_f32(false, a, false, b,
                                               (short)0, c0, false, false);
    a.x = ra[u0 + 16]; a.y = rb[u0 + 16];
    c1 = __builtin_amdgcn_wmma_f32_16x16x4_f32(false, a, false, b,
                                               (short)0, c1, false, false);
    a.x = ra[u0 + 32]; a.y = rb[u0 + 32];
    c2 = __builtin_amdgcn_wmma_f32_16x16x4_f32(false, a, false, b,
                                               (short)0, c2, false, false);
    a.x = ra[u0 + 48]; a.y = rb[u0 + 48];
    c3 = __builtin_amdgcn_wmma_f32_16x16x4_f32(false, a, false, b,
                                               (short)0, c3, false, false);

    // ---- K-chunk B: nonzeros k0+4..k0+7 ----
    b.x = hiHalf ? v6 : v4;
    b.y = hiHalf ? v7 : v5;
    ra = Xt + (size_t)(hiHalf ? j6 : j4) * N_USERS;
    rb = Xt + (size_t)(hiHalf ? j7 : j5) * N_USERS;

    a.x = ra[u0];      a.y = rb[u0];
    c0 = __builtin_amdgcn_wmma_f32_16x16x4_f32(false, a, false, b,
                                               (short)0, c0, false, false);
    a.x = ra[u0 + 16]; a.y = rb[u0 + 16];
    c1 = __builtin_amdgcn_wmma_f32_16x16x4_f32(false, a, false, b,
                                               (short)0, c1, false, false);
    a.x = ra[u0 + 32]; a.y = rb[u0 + 32];
    c2 = __builtin_amdgcn_wmma_f32_16x16x4_f32(false, a, false, b,
                                               (short)0, c2, false, false);
    a.x = ra[u0 + 48]; a.y = rb[u0 + 48];
    c3 = __builtin_amdgcn_wmma_f32_16x16x4_f32(false, a, false, b,
                                               (short)0, c3, false, false);
  }

  // Extract column N=0: lane 0 -> users M=0..7, lane 16 -> users M=8..15.
  if (lane == 0 || lane == 16) {
    const int ub  = wave * 64;
    const int off = hiHalf ? 8 : 0;
#pragma unroll
    for (int r = 0; r < 8; ++r) {
      out[(size_t)(ub +  0 + off + r) * N_ITEMS + i] = c0[r];
      out[(size_t)(ub + 16 + off + r) * N_ITEMS + i] = c1[r];
      out[(size_t)(ub + 32 + off + r) * N_ITEMS + i] = c2[r];
      out[(size_t)(ub + 48 + off + r) * N_ITEMS + i] = c3[r];
    }
  }
}

// ---------------------------------------------------------------------------
extern "C" void kernel_launch(void* const* d_in, const int* in_sizes, int n_in,
                              void* d_out, int out_size, void* d_ws, size_t ws_size,
                              hipStream_t stream) {
  const float* X         = (const float*)d_in[0];  // [512, 20000] f32
  const float* S_values  = (const float*)d_in[1];  // [4M] f32
  const int*   S_indices = (const int*)d_in[2];    // [4M, 2] i32
  float* out = (float*)d_out;                      // [512, 20000] f32

  // Workspace layout (~74.5 MB total):
  float*    Xt        = (float*)d_ws;                        // 20000*512 f32
  float*    csr_val   = Xt + (size_t)N_ITEMS * N_USERS;      // PAD_MAX f32
  int*      csr_col   = (int*)(csr_val + PAD_MAX);           // PAD_MAX i32
  unsigned* row_start = (unsigned*)(csr_col + PAD_MAX);      // N_ITEMS+1 u32
  unsigned* cursor    = row_start + (N_ITEMS + 1);           // N_ITEMS u32
  unsigned* counts    = cursor + N_ITEMS;                    // N_ITEMS u32

  lira_zero_counts<<<(N_ITEMS + 255) / 256, 256, 0, stream>>>(counts, N_ITEMS);
  lira_zero_csr<<<(PAD_MAX + 255) / 256, 256, 0, stream>>>(csr_val, csr_col);
  lira_hist<<<NNZ_TOTAL / 256, 256, 0, stream>>>(S_indices, counts);
  lira_scan<<<1, 1024, 0, stream>>>(counts, row_start, cursor);
  lira_scatter<<<NNZ_TOTAL / 256, 256, 0, stream>>>(S_values, S_indices, cursor,
                                                    csr_val, csr_col);
  lira_transpose<<<dim3(N_ITEMS / 32, N_USERS / 32), dim3(32, 8), 0, stream>>>(X, Xt);
  lira_spmm_wmma<<<N_ITEMS, 256, 0, stream>>>(Xt, csr_val, csr_col, row_start, out);
}